// GOGlobalEmbedder_89421219103620
// MI455X (gfx1250) — compile-verified
//
#include <hip/hip_runtime.h>
#include <math.h>

// ---------------------------------------------------------------------------
// Types for CDNA5 WMMA (gfx1250, wave32)
// ---------------------------------------------------------------------------
typedef float v2f __attribute__((ext_vector_type(2)));
typedef float v8f __attribute__((ext_vector_type(8)));

// 16-byte chunk type + address-space-qualified pointers for async-to-LDS
typedef int v4i __attribute__((__vector_size__(4 * sizeof(int))));
typedef __attribute__((address_space(1))) v4i* global_v4i_ptr;
typedef __attribute__((address_space(3))) v4i* lds_v4i_ptr;

#define WARP 32
#define NEG_SLOPE 0.2f
#define LDS_PAD 4   // keep rows 16B-aligned for async B128 copies; banks stay distinct

#if __has_builtin(__builtin_amdgcn_global_load_async_to_lds_b128) && \
    __has_builtin(__builtin_amdgcn_s_wait_asynccnt)
#define USE_ASYNC_LDS 1
#else
#define USE_ASYNC_LDS 0
#endif

// ---------------------------------------------------------------------------
// fp32 GEMM via V_WMMA_F32_16X16X4_F32.
// C[M,Ncols] = A[M,K] @ B[K,Ncols], all row-major fp32.
// Block = 512 threads = 16 waves. Each block owns a 16-row strip of A
// (staged in padded LDS, async-to-LDS when available); each wave computes
// one or more 16x16 output tiles.
// ---------------------------------------------------------------------------
__global__ void gemm_f32_wmma_kernel(const float* __restrict__ A,
                                     const float* __restrict__ B,
                                     float* __restrict__ C,
                                     int M, int K, int Ncols) {
    extern __shared__ float ldsA[];        // 16 * (K+LDS_PAD) floats
    const int tid   = threadIdx.x;
    const int lane  = tid & (WARP - 1);
    const int wave  = tid >> 5;
    const int nWaves = blockDim.x >> 5;
    const int row0  = blockIdx.x * 16;
    const int ldk   = K + LDS_PAD;         // padded LDS stride

#if USE_ASYNC_LDS
    // Stage the 16xK A strip with GLOBAL_LOAD_ASYNC_TO_LDS_B128 (16B per lane,
    // bypasses VGPRs, tracked by ASYNCcnt).
    {
        const int cpr    = K >> 2;         // 16B chunks per row
        const int chunks = 16 * cpr;
        for (int i = tid; i < chunks; i += blockDim.x) {
            const int r = i / cpr;
            const int c = i - r * cpr;
            const float* gsrc = A + (size_t)(row0 + r) * K + (size_t)c * 4;
            float* ldst = &ldsA[r * ldk + c * 4];
            __builtin_amdgcn_global_load_async_to_lds_b128(
                (global_v4i_ptr)gsrc, (lds_v4i_ptr)ldst, 0, 0);
        }
        __builtin_amdgcn_s_wait_asynccnt(0);
    }
#else
    for (int i = tid; i < 16 * K; i += blockDim.x) {
        int r = i / K;
        int k = i - r * K;
        ldsA[r * ldk + k] = A[(size_t)(row0 + r) * K + k];
    }
#endif
    __syncthreads();

    const int mrow = lane & 15;            // A row within tile / B,C column within tile
    const int kgrp = (lane >> 4) * 2;      // K sub-offset per half-wave (ISA 16x4 f32 layout)
    const int ntiles = Ncols >> 4;

    for (int nt = wave; nt < ntiles; nt += nWaves) {
        const int ncol = nt * 16 + mrow;
        v8f acc = {0.f, 0.f, 0.f, 0.f, 0.f, 0.f, 0.f, 0.f};
        for (int k = 0; k < K; k += 4) {
            const int kk = k + kgrp;
            v2f a, b;
            a.x = ldsA[mrow * ldk + kk];
            a.y = ldsA[mrow * ldk + kk + 1];
            b.x = B[(size_t)kk * Ncols + ncol];
            b.y = B[(size_t)(kk + 1) * Ncols + ncol];
            // D = A(16x4) x B(4x16) + C, fp32 matrix pipe
            acc = __builtin_amdgcn_wmma_f32_16x16x4_f32(
                false, a, false, b, (short)0, acc, false, false);
        }
        // C/D layout: VGPR v -> row (v + 8*(lane/16)), col = lane%16
        const int rbase = row0 + (lane >> 4) * 8;
#pragma unroll
        for (int v = 0; v < 8; ++v)
            C[(size_t)(rbase + v) * Ncols + ncol] = acc[v];
    }
}

// ---------------------------------------------------------------------------
// Per-(node,head) attention scores: e_src = <h, a_src>, e_dst = <h, a_dst>.
// One wave32 per (node,head); butterfly reduction across lanes.
// ---------------------------------------------------------------------------
__global__ void gat_scores_kernel(const float* __restrict__ h,
                                  const float* __restrict__ a_src,
                                  const float* __restrict__ a_dst,
                                  float* __restrict__ e_src,
                                  float* __restrict__ e_dst,
                                  int N, int H, int C) {
    const int warp = (int)((blockIdx.x * blockDim.x + threadIdx.x) >> 5);
    const int lane = threadIdx.x & (WARP - 1);
    if (warp >= N * H) return;
    const int node = warp / H;
    const int head = warp - node * H;
    const float* hp = h + (size_t)node * H * C + (size_t)head * C;
    const float* as = a_src + head * C;
    const float* ad = a_dst + head * C;
    float ss = 0.f, sd = 0.f;
    for (int c = lane; c < C; c += WARP) {
        float v = hp[c];
        ss += v * as[c];
        sd += v * ad[c];
    }
#pragma unroll
    for (int off = 16; off > 0; off >>= 1) {
        ss += __shfl_xor(ss, off, WARP);
        sd += __shfl_xor(sd, off, WARP);
    }
    if (lane == 0) {
        e_src[warp] = ss;
        e_dst[warp] = sd;
    }
}

// ---------------------------------------------------------------------------
// Edge helpers: edges [0,E) come from edge_index, edges [E,E+N) are self loops.
// ---------------------------------------------------------------------------
__device__ __forceinline__ void edge_nodes(int e, const int* __restrict__ ei,
                                           int E, int& s, int& d) {
    if (e < E) { s = ei[e]; d = ei[E + e]; }
    else       { s = d = e - E; }
}

// Pass 1: segment max of leaky_relu(e_src[s]+e_dst[d]) per (dst, head).
__global__ void edge_max_kernel(const int* __restrict__ ei, int E, int Etot,
                                const float* __restrict__ e_src,
                                const float* __restrict__ e_dst,
                                float* __restrict__ m, int H) {
    const int idx = blockIdx.x * blockDim.x + threadIdx.x;
    if (idx >= Etot * H) return;
    const int e = idx / H;
    const int h = idx - e * H;
    int s, d; edge_nodes(e, ei, E, s, d);
    float sc = e_src[s * H + h] + e_dst[d * H + h];
    sc = (sc > 0.f) ? sc : NEG_SLOPE * sc;
    atomicMax(&m[d * H + h], sc);   // GLOBAL_ATOMIC_MAX_NUM_F32
}

// Pass 2: numerator exp(e - m[dst]) stored per edge + segment-sum denominator.
__global__ void edge_exp_kernel(const int* __restrict__ ei, int E, int Etot,
                                const float* __restrict__ e_src,
                                const float* __restrict__ e_dst,
                                const float* __restrict__ m,
                                float* __restrict__ ex,
                                float* __restrict__ den, int H) {
    const int idx = blockIdx.x * blockDim.x + threadIdx.x;
    if (idx >= Etot * H) return;
    const int e = idx / H;
    const int h = idx - e * H;
    int s, d; edge_nodes(e, ei, E, s, d);
    float sc = e_src[s * H + h] + e_dst[d * H + h];
    sc = (sc > 0.f) ? sc : NEG_SLOPE * sc;
    float mx = m[d * H + h];
    if (!isfinite(mx)) mx = 0.f;
    const float v = expf(sc - mx);
    ex[idx] = v;
    atomicAdd(&den[d * H + h], v);  // GLOBAL_ATOMIC_ADD_F32
}

// Pass 3: weighted aggregation agg[dst] += h[src] * alpha. One wave per edge.
__global__ void edge_agg_kernel(const int* __restrict__ ei, int E, int Etot,
                                const float* __restrict__ hfeat,
                                const float* __restrict__ ex,
                                const float* __restrict__ den,
                                float* __restrict__ agg, int H, int C) {
    const int warp = (int)((blockIdx.x * blockDim.x + threadIdx.x) >> 5);
    const int lane = threadIdx.x & (WARP - 1);
    if (warp >= Etot) return;
    int s, d; edge_nodes(warp, ei, E, s, d);
    const size_t sbase = (size_t)s * H * C;
    const size_t dbase = (size_t)d * H * C;
    for (int h = 0; h < H; ++h) {
        const float alpha = ex[(size_t)warp * H + h] / (den[d * H + h] + 1e-16f);
        for (int c = lane; c < C; c += WARP)
            atomicAdd(&agg[dbase + (size_t)h * C + c],
                      hfeat[sbase + (size_t)h * C + c] * alpha);
    }
}

// agg = relu(agg + bias)   (layer-1 epilogue, feeds GEMM2)
__global__ void bias_relu_kernel(float* __restrict__ x, const float* __restrict__ b,
                                 int total, int F) {
    const int i = blockIdx.x * blockDim.x + threadIdx.x;
    if (i >= total) return;
    const float v = x[i] + b[i % F];
    x[i] = v > 0.f ? v : 0.f;
}

// mean over heads + bias + LayerNorm. One wave per node, C = 128 (4/lane).
__global__ void finalize_kernel(const float* __restrict__ agg2,
                                const float* __restrict__ b2,
                                const float* __restrict__ gamma,
                                const float* __restrict__ beta,
                                float* __restrict__ out, int N, int H, int C) {
    const int warp = (int)((blockIdx.x * blockDim.x + threadIdx.x) >> 5);
    const int lane = threadIdx.x & (WARP - 1);
    if (warp >= N) return;
    const float* base = agg2 + (size_t)warp * H * C;
    const float invH = 1.f / (float)H;
    float v[4];
    float s = 0.f;
#pragma unroll
    for (int i = 0; i < 4; ++i) {
        const int c = lane + i * WARP;
        float acc = 0.f;
        for (int h = 0; h < H; ++h) acc += base[h * C + c];
        v[i] = acc * invH + b2[c];
        s += v[i];
    }
#pragma unroll
    for (int off = 16; off > 0; off >>= 1) s += __shfl_xor(s, off, WARP);
    const float mu = s / (float)C;
    float q = 0.f;
#pragma unroll
    for (int i = 0; i < 4; ++i) { const float dv = v[i] - mu; q += dv * dv; }
#pragma unroll
    for (int off = 16; off > 0; off >>= 1) q += __shfl_xor(q, off, WARP);
    const float rs = rsqrtf(q / (float)C + 1e-5f);
#pragma unroll
    for (int i = 0; i < 4; ++i) {
        const int c = lane + i * WARP;
        out[(size_t)warp * C + c] = (v[i] - mu) * rs * gamma[c] + beta[c];
    }
}

__global__ void fill_kernel(float* __restrict__ p, float v, size_t n) {
    const size_t i = blockIdx.x * (size_t)blockDim.x + threadIdx.x;
    if (i < n) p[i] = v;
}

// ---------------------------------------------------------------------------
// Host launcher
// ---------------------------------------------------------------------------
static inline unsigned cdiv(size_t a, size_t b) { return (unsigned)((a + b - 1) / b); }

extern "C" void kernel_launch(void* const* d_in, const int* in_sizes, int n_in,
                              void* d_out, int out_size, void* d_ws, size_t ws_size,
                              hipStream_t stream) {
    const int IN_DIM = 128, HID = 64, OUT = 128, H = 4;
    const int F1 = H * HID;   // 256
    const int F2 = H * OUT;   // 512

    const float* x      = (const float*)d_in[0];
    const int*   ei     = (const int*)d_in[1];
    const float* W1     = (const float*)d_in[2];
    const float* a_src1 = (const float*)d_in[3];
    const float* a_dst1 = (const float*)d_in[4];
    const float* b1     = (const float*)d_in[5];
    const float* W2     = (const float*)d_in[6];
    const float* a_src2 = (const float*)d_in[7];
    const float* a_dst2 = (const float*)d_in[8];
    const float* b2     = (const float*)d_in[9];
    const float* gamma  = (const float*)d_in[10];
    const float* beta   = (const float*)d_in[11];
    float* out = (float*)d_out;

    const int N    = in_sizes[0] / IN_DIM;  // 50000
    const int E    = in_sizes[1] / 2;       // 800000
    const int Etot = E + N;                 // + self loops

    // carve workspace
    float* ws = (float*)d_ws;
    size_t off = 0;
    auto carve = [&](size_t n) { float* p = ws + off; off += (n + 63) & ~(size_t)63; return p; };
    float* h1   = carve((size_t)N * F1);
    float* agg1 = carve((size_t)N * F1);
    float* h2   = carve((size_t)N * F2);
    float* agg2 = carve((size_t)N * F2);
    float* es1  = carve((size_t)N * H);
    float* ed1  = carve((size_t)N * H);
    float* m1   = carve((size_t)N * H);
    float* dn1  = carve((size_t)N * H);
    float* es2  = carve((size_t)N * H);
    float* ed2  = carve((size_t)N * H);
    float* m2   = carve((size_t)N * H);
    float* dn2  = carve((size_t)N * H);
    float* ex   = carve((size_t)Etot * H);  // reused across layers
    (void)ws_size; (void)n_in; (void)out_size;

    const float NEG_INF = -__builtin_huge_valf();

    // ---------------- Layer 1 ----------------
    gemm_f32_wmma_kernel<<<N / 16, 512, 16 * (IN_DIM + LDS_PAD) * sizeof(float), stream>>>(
        x, W1, h1, N, IN_DIM, F1);

    gat_scores_kernel<<<cdiv((size_t)N * H * WARP, 256), 256, 0, stream>>>(
        h1, a_src1, a_dst1, es1, ed1, N, H, HID);

    fill_kernel<<<cdiv((size_t)N * H, 256), 256, 0, stream>>>(m1, NEG_INF, (size_t)N * H);
    fill_kernel<<<cdiv((size_t)N * H, 256), 256, 0, stream>>>(dn1, 0.f, (size_t)N * H);
    fill_kernel<<<cdiv((size_t)N * F1, 256), 256, 0, stream>>>(agg1, 0.f, (size_t)N * F1);

    edge_max_kernel<<<cdiv((size_t)Etot * H, 256), 256, 0, stream>>>(
        ei, E, Etot, es1, ed1, m1, H);
    edge_exp_kernel<<<cdiv((size_t)Etot * H, 256), 256, 0, stream>>>(
        ei, E, Etot, es1, ed1, m1, ex, dn1, H);
    edge_agg_kernel<<<cdiv((size_t)Etot * WARP, 256), 256, 0, stream>>>(
        ei, E, Etot, h1, ex, dn1, agg1, H, HID);

    bias_relu_kernel<<<cdiv((size_t)N * F1, 256), 256, 0, stream>>>(
        agg1, b1, N * F1, F1);

    // ---------------- Layer 2 ----------------
    gemm_f32_wmma_kernel<<<N / 16, 512, 16 * (F1 + LDS_PAD) * sizeof(float), stream>>>(
        agg1, W2, h2, N, F1, F2);

    gat_scores_kernel<<<cdiv((size_t)N * H * WARP, 256), 256, 0, stream>>>(
        h2, a_src2, a_dst2, es2, ed2, N, H, OUT);

    fill_kernel<<<cdiv((size_t)N * H, 256), 256, 0, stream>>>(m2, NEG_INF, (size_t)N * H);
    fill_kernel<<<cdiv((size_t)N * H, 256), 256, 0, stream>>>(dn2, 0.f, (size_t)N * H);
    fill_kernel<<<cdiv((size_t)N * F2, 256), 256, 0, stream>>>(agg2, 0.f, (size_t)N * F2);

    edge_max_kernel<<<cdiv((size_t)Etot * H, 256), 256, 0, stream>>>(
        ei, E, Etot, es2, ed2, m2, H);
    edge_exp_kernel<<<cdiv((size_t)Etot * H, 256), 256, 0, stream>>>(
        ei, E, Etot, es2, ed2, m2, ex, dn2, H);
    edge_agg_kernel<<<cdiv((size_t)Etot * WARP, 256), 256, 0, stream>>>(
        ei, E, Etot, h2, ex, dn2, agg2, H, OUT);

    finalize_kernel<<<cdiv((size_t)N * WARP, 256), 256, 0, stream>>>(
        agg2, b2, gamma, beta, out, N, H, OUT);
}